// FrustumEncoder_3358664425622
// MI455X (gfx1250) — compile-verified
//
#include <hip/hip_runtime.h>
#include <math.h>

#define NPOINTS 2000000
#define NFR     14400
#define NCH     8
#define TILE    16
#define NTILES  (NPOINTS / TILE)   // 125000 exactly

typedef __attribute__((ext_vector_type(2))) float    v2f;
typedef __attribute__((ext_vector_type(8))) float    v8f;
typedef __attribute__((ext_vector_type(4))) unsigned v4u;
typedef __attribute__((ext_vector_type(4))) int      v4i;
typedef __attribute__((ext_vector_type(8))) int      v8i;

#if __has_builtin(__builtin_amdgcn_tensor_load_to_lds) && \
    __has_builtin(__builtin_amdgcn_s_wait_tensorcnt)
#define HAVE_TDM 1
#else
#define HAVE_TDM 0
#endif

struct alignas(16) WaveLds {
    float tile[2][TILE * NCH];  // double-buffered 128-f32 tile (pc or dev)
    int   rank[TILE];           // segment rank of each point in tile
    int   frOfRank[TILE];       // frustum id for each rank
    int   maxRank;
    int   pad3[3];
};

#if HAVE_TDM
// One 1-D TDM descriptor: copy 512 B (128 x f32) from wave-uniform global
// address into LDS at ldsAddr.  D# layout per CDNA5 ISA §8 (count=1, type=2,
// data_size=4B, tensor_dim0=tile_dim0=128, no padding/iterate/multicast).
__device__ __forceinline__ void tdm_load_tile(const float* gsrc, unsigned ldsAddr) {
    unsigned long long ga = (unsigned long long)(uintptr_t)gsrc;
    v4u g0 = { 1u,                                   // count=1, user mode
               ldsAddr,                              // lds_addr (bytes)
               (unsigned)ga,                         // global_addr[31:0]
               (unsigned)((ga >> 32) & 0x1FFFFFFull) // global_addr[56:32]
                   | (2u << 30) };                   // type=2 (image)
    v8i g1 = { (int)0x00020000,   // mask=0, data_size=2 (4 bytes)
               (int)(128u << 16), // tensor_dim0 = 128 (bits 48..63)
               (int)(1u  << 16),  // tensor_dim1 = 1   (bits 80..95)
               (int)(128u << 16), // tile_dim0  = 128  (bits 112..127)
               0,                 // tile_dim1 = 0 (unused), tile_dim2 = 0
               128,               // tensor_dim0_stride = 128
               0, 0 };
    v4i gz = { 0, 0, 0, 0 };
#if defined(__clang_major__) && __clang_major__ >= 23
    v8i gz8 = { 0, 0, 0, 0, 0, 0, 0, 0 };
    __builtin_amdgcn_tensor_load_to_lds(g0, g1, gz, gz, gz8, 0);
#else
    __builtin_amdgcn_tensor_load_to_lds(g0, g1, gz, gz, 0);
#endif
}
#endif

// ---------------- zero init of mean/std/max stat regions -------------------
__global__ __launch_bounds__(256) void k_zero(float* __restrict__ p, int n) {
    int i = blockIdx.x * blockDim.x + threadIdx.x;
    if (i < n) p[i] = 0.0f;
}

// ---------------- pass1 (sums+max) / pass2 (var sums), WMMA aggregated -----
template <bool PASS2>
__global__ __launch_bounds__(256)
void k_segsum(const float* __restrict__ pc, const int* __restrict__ ifr,
              const int* __restrict__ iinv, const float* __restrict__ mean,
              float* __restrict__ accum, float* __restrict__ omax)
{
    __shared__ WaveLds lds[8];
    const int lane = threadIdx.x & 31;
    const int wv   = threadIdx.x >> 5;
    WaveLds& W = lds[wv];
    const int gw = (blockIdx.x * blockDim.x + threadIdx.x) >> 5;
    const int nW = (gridDim.x * blockDim.x) >> 5;
    const int m  = lane & 15;   // WMMA M row / point index for lanes 0..15
    const int hi = lane >> 4;

    int cur = 0;
#if HAVE_TDM
    const unsigned ldsBuf[2] = { (unsigned)(uintptr_t)&W.tile[0][0],
                                 (unsigned)(uintptr_t)&W.tile[1][0] };
    if (!PASS2 && gw < NTILES)                  // prologue: prefetch first tile
        tdm_load_tile(pc + (size_t)gw * TILE * NCH, ldsBuf[0]);
#endif

    for (int t = gw; t < NTILES; t += nW) {
        const int p0 = t * TILE;

        if (PASS2) {
            // dev = pc - mean[i_frustum[i_inv[p]]]  staged through VGPRs
            if (t + nW < NTILES)
                __builtin_prefetch(pc + (size_t)(t + nW) * TILE * NCH, 0, 0);
            float4 pv = ((const float4*)(pc + (size_t)p0 * NCH))[lane];
            const int p   = lane >> 1;          // point for this lane's 4 ch
            const int fid = ifr[iinv[p0 + p]];
            const float4 mv = ((const float4*)(mean + fid * NCH))[lane & 1];
            pv.x -= mv.x; pv.y -= mv.y; pv.z -= mv.z; pv.w -= mv.w;
            ((float4*)W.tile[0])[lane] = pv;
        } else {
#if HAVE_TDM
            // TDM double buffer: kick DMA for next tile, wait for current
            if (t + nW < NTILES) {
                tdm_load_tile(pc + (size_t)(t + nW) * TILE * NCH, ldsBuf[cur ^ 1]);
                __builtin_amdgcn_s_wait_tensorcnt(1);
            } else {
                __builtin_amdgcn_s_wait_tensorcnt(0);
            }
            asm volatile("" ::: "memory");
#else
            float4 pv = ((const float4*)(pc + (size_t)p0 * NCH))[lane];
            ((float4*)W.tile[0])[lane] = pv;
#endif
        }
        const float* tptr = W.tile[PASS2 ? 0 : cur];

        // ---- frustum ids + rank (number of distinct segments before point) ----
        const int fr     = ifr[p0 + m];         // scatter key: plain i_frustum
        const int frPrev = __shfl_up(fr, 1, 32);
        const int flag   = (lane == 0 || lane >= 16) ? 0 : (fr != frPrev ? 1 : 0);
        int rank = flag;
        #pragma unroll
        for (int off = 1; off < 16; off <<= 1) {
            int o = __shfl_up(rank, off, 32);
            if (lane >= off) rank += o;
        }
        if (lane < 16) {
            W.rank[lane] = rank;
            if (flag || lane == 0) W.frOfRank[rank] = fr;
            if (lane == 15)        W.maxRank = rank;
        }
        __builtin_amdgcn_wave_barrier();   // keep LDS writes before reads

        // ---- segment-sum via one-hot WMMA: C[rank][ch] = sum of tile rows ----
        // A (16x4 f32): lanes0-15 M=0..15 {K0,K1}; lanes16-31 {K2,K3}
        // B (4x16 f32): mirrored K assignment, N striped across lanes
        v8f acc = {};
        #pragma unroll
        for (int kk = 0; kk < 4; ++kk) {
            const int k0 = kk * 4 + hi * 2;
            const int r0 = W.rank[k0];
            const int r1 = W.rank[k0 + 1];
            v2f a; a.x = (r0 == m) ? 1.0f : 0.0f;
                   a.y = (r1 == m) ? 1.0f : 0.0f;
            const int ch = m & 7;          // cols 8..15 mirror 0..7 (ignored)
            float bx = tptr[k0 * NCH + ch];
            float by = tptr[(k0 + 1) * NCH + ch];
            if (PASS2) { bx *= bx; by *= by; }   // dev*dev for variance sums
            v2f b; b.x = bx; b.y = by;
            acc = __builtin_amdgcn_wmma_f32_16x16x4_f32(
                false, a, false, b, (short)0, acc, false, false);
        }
        const int maxRank = W.maxRank;

        // ---- scatter valid rank rows (typically ~1.1 per tile) to L2 ----
        #pragma unroll
        for (int j = 0; j < 8; ++j) {
            const int row = hi * 8 + j;    // C layout: VGPR j = rows j / j+8
            const int chn = lane & 15;
            if (chn < 8 && row <= maxRank) {
                const int fdst = W.frOfRank[row];
                atomicAdd(&accum[fdst * NCH + chn], acc[j]);
            }
        }

        if (!PASS2) {
            // ---- segment max: segmented shuffle scan, atomics at segment end ----
            float v[8];
            #pragma unroll
            for (int c = 0; c < 8; ++c) v[c] = tptr[m * NCH + c];
            #pragma unroll
            for (int off = 1; off < 16; off <<= 1) {
                const int of = __shfl_up(fr, off, 32);
                #pragma unroll
                for (int c = 0; c < 8; ++c) {
                    float ov = __shfl_up(v[c], off, 32);
                    if (lane >= off && lane < 16 && of == fr) v[c] = fmaxf(v[c], ov);
                }
            }
            const int frN = __shfl_down(fr, 1, 32);
            const bool segEnd = (lane < 16) && (lane == 15 || frN != fr);
            if (segEnd) {
                #pragma unroll
                for (int c = 0; c < 8; ++c)
                    __hip_atomic_fetch_max(&omax[fr * NCH + c], v[c],
                                           __ATOMIC_RELAXED, __HIP_MEMORY_SCOPE_AGENT);
            }
            cur ^= 1;
        }
    }
}

// ---------------- finalize mean in place: mean = where(cnt!=0, s/cnt, s) ----
__global__ __launch_bounds__(256)
void k_fin_mean(float* __restrict__ sums, const float* __restrict__ cnts) {
    int i = blockIdx.x * blockDim.x + threadIdx.x;
    if (i >= NFR * NCH) return;
    const float c = cnts[i >> 3];
    const float s = sums[i];
    sums[i] = (c != 0.0f) ? s / fmaxf(c, 1.0f) : s;
}

// ---------------- finalize std in place ------------------------------------
__global__ __launch_bounds__(256)
void k_fin_std(float* __restrict__ vs, const float* __restrict__ cnts) {
    int i = blockIdx.x * blockDim.x + threadIdx.x;
    if (i >= NFR * NCH) return;
    const float c = cnts[i >> 3];
    float v = vs[i];
    v = (c != 0.0f) ? v / fmaxf(c, 1.0f) : v;
    vs[i] = (v > 0.0f) ? sqrtf(v) : v;
}

// ---------------- pc_norm = (pc - mean_g) / where(std_g>0, std_g, 1) -------
__global__ __launch_bounds__(256)
void k_norm(const float* __restrict__ pc, const int* __restrict__ ifr,
            const int* __restrict__ iinv, const float* __restrict__ mean,
            const float* __restrict__ stdv, float* __restrict__ out)
{
    int tid = blockIdx.x * blockDim.x + threadIdx.x;
    if (tid >= NPOINTS * NCH / 4) return;
    const int p    = tid >> 1;
    const int half = tid & 1;
    const int fid  = ifr[iinv[p]];
    const float4 x  = ((const float4*)pc)[tid];
    const float4 mv = ((const float4*)(mean + fid * NCH))[half];
    const float4 sv = ((const float4*)(stdv + fid * NCH))[half];
    float4 o;
    o.x = (x.x - mv.x) / (sv.x > 0.0f ? sv.x : 1.0f);
    o.y = (x.y - mv.y) / (sv.y > 0.0f ? sv.y : 1.0f);
    o.z = (x.z - mv.z) / (sv.z > 0.0f ? sv.z : 1.0f);
    o.w = (x.w - mv.w) / (sv.w > 0.0f ? sv.w : 1.0f);
    ((float4*)out)[tid] = o;
}

extern "C" void kernel_launch(void* const* d_in, const int* in_sizes, int n_in,
                              void* d_out, int out_size, void* d_ws, size_t ws_size,
                              hipStream_t stream) {
    const float* pc   = (const float*)d_in[0];
    const int*   ifr  = (const int*)d_in[1];
    const int*   iinv = (const int*)d_in[2];
    const float* cnts = (const float*)d_in[3];

    // d_out layout: mean[F*C] | std[F*C] | max[F*C] | pc_norm[N*C]
    float* mean  = (float*)d_out;          // doubles as sum accumulator
    float* stdv  = mean + NFR * NCH;       // doubles as var-sum accumulator
    float* omax  = stdv + NFR * NCH;
    float* pnorm = omax + NFR * NCH;

    const int statN = 3 * NFR * NCH;
    k_zero<<<(statN + 255) / 256, 256, 0, stream>>>(mean, statN);

    k_segsum<false><<<1024, 256, 0, stream>>>(pc, ifr, nullptr, nullptr, mean, omax);
    k_fin_mean<<<(NFR * NCH + 255) / 256, 256, 0, stream>>>(mean, cnts);
    k_segsum<true><<<1024, 256, 0, stream>>>(pc, ifr, iinv, mean, stdv, nullptr);
    k_fin_std<<<(NFR * NCH + 255) / 256, 256, 0, stream>>>(stdv, cnts);

    const int nrm = NPOINTS * NCH / 4;
    k_norm<<<(nrm + 255) / 256, 256, 0, stream>>>(pc, ifr, iinv, mean, stdv, pnorm);
}